// QuantumVariationalLayer_57690000720065
// MI455X (gfx1250) — compile-verified
//
#include <hip/hip_runtime.h>
#include <math.h>

// ---------------------------------------------------------------------------
// 14-qubit variational circuit simulator for MI455X (gfx1250, wave32).
//
// One workgroup (256 threads = 8 waves) per batch sample; the 2^14 complex64
// state vector (128 KB) lives in LDS. Every single-qubit gate (RX embedding
// and Rot) runs on the matrix pipe via V_WMMA_F32_16X16X4_F32: the 2x2
// complex gate is recast as a real 4x4 matrix G acting on [re0,im0,re1,im1];
// A(16x4) holds G replicated 4x in M, B(4x16) holds 16 amplitude pairs (one
// ds_load_b64 per lane), D(16x16) returns the transformed pairs (one
// ds_store_b64 per lane). fp32 WMMA is mandatory: the reference state is
// complex64 and 98 sequential gates would destroy fp16/fp8 precision.
//
// The pair loop is a counted scalar loop (64 trips, SALU-controlled) so EXEC
// is never modified around the WMMA, per the ISA's EXEC==all-ones rule.
// ---------------------------------------------------------------------------

typedef __attribute__((ext_vector_type(2))) float v2f;
typedef __attribute__((ext_vector_type(8))) float v8f;

#define NQ        14
#define NL        6
#define NSTATES   16384      // 2^14
#define NPAIRS    8192       // NSTATES / 2
#define NTHREADS  256
#define NWAVES    8          // wave32: 256/32
#define PAIR_ITERS (NPAIRS / (16 * NWAVES))   // 64

// Apply complex 2x2 gate [[u00, u01],[u10, u11]] on `wire` (wire 0 = MSB).
// psi is the LDS state as interleaved (re, im) float2.
__device__ __forceinline__ void apply_1q_wmma(
    float2* __restrict__ psi, int wire,
    float u00r, float u00i, float u01r, float u01i,
    float u10r, float u10i, float u11r, float u11i)
{
    const unsigned tid  = threadIdx.x;
    const unsigned lane = tid & 31u;
    const unsigned wv   = tid >> 5;       // wave id 0..7
    const unsigned h    = lane >> 4;      // 0: handles i0 half, 1: handles i1 half
    const unsigned r    = lane & 3u;      // which row of the real 4x4 gate
    const int      shift = 13 - wire;
    const unsigned post  = 1u << shift;

    // Real 4x4 matrix G acting on [re0, im0, re1, im1]:
    //  row0 (re0'): [ u00r, -u00i,  u01r, -u01i ]
    //  row1 (im0'): [ u00i,  u00r,  u01i,  u01r ]
    //  row2 (re1'): [ u10r, -u10i,  u11r, -u11i ]
    //  row3 (im1'): [ u10i,  u10r,  u11i,  u11r ]
    float g0, g1, g2, g3;
    if (r == 0u)      { g0 =  u00r; g1 = -u00i; g2 =  u01r; g3 = -u01i; }
    else if (r == 1u) { g0 =  u00i; g1 =  u00r; g2 =  u01i; g3 =  u01r; }
    else if (r == 2u) { g0 =  u10r; g1 = -u10i; g2 =  u11r; g3 = -u11i; }
    else              { g0 =  u10i; g1 =  u10r; g2 =  u11i; g3 =  u11r; }

    // A-matrix layout (f32 16x4): lane L holds A[M=L%16][K=2*(L/16)+j] in vgpr j.
    // A rows m = G[m & 3]  (gate replicated over the 4 row-blocks).
    v2f A;
    A.x = h ? g2 : g0;   // K = 2h
    A.y = h ? g3 : g1;   // K = 2h + 1

    // Per-lane base pair index and its (i0-half) address offset are invariant.
    const unsigned pcol = wv * 16u + (lane & 15u);

    // 8192 pairs, 16 per WMMA, round-robined over 8 waves -> 64 counted trips.
    // Scalar loop control => EXEC untouched around the WMMA.
#pragma unroll 2
    for (int it = 0; it < PAIR_ITERS; ++it) {
        const unsigned p    = pcol + (unsigned)it * (16u * NWAVES);
        const unsigned i0   = ((p >> shift) << (shift + 1)) | (p & (post - 1u));
        const unsigned addr = i0 + (h ? post : 0u);

        // B-matrix layout (f32 4x16): lanes 0-15 hold K=0,1 (vgpr 0,1),
        // lanes 16-31 hold K=2,3. Column N = lane%16 = pair index.
        // One ds_load_b64: (re,im) of this lane's half of the pair.
        const float2 vin = psi[addr];
        v2f B; B.x = vin.x; B.y = vin.y;

        v8f C = {};  // pure transform, no accumulation
        v8f D = __builtin_amdgcn_wmma_f32_16x16x4_f32(
            /*neg_a=*/false, A, /*neg_b=*/false, B,
            /*c_mod=*/(short)0, C, /*reuse_a=*/false, /*reuse_b=*/false);

        // D layout: lane L holds D[M=vgpr+8*(L/16)][N=L%16].
        // lane<16 : d[0],d[1] = rows 0,1  = (re0',im0') -> store at i0
        // lane>=16: d[2],d[3] = rows 10,11 ≡ G rows 2,3 = (re1',im1') -> store at i1
        float2 vout;
        vout.x = h ? D[2] : D[0];
        vout.y = h ? D[3] : D[1];
        psi[addr] = vout;
    }
}

__global__ __launch_bounds__(NTHREADS)
void qvl_qsim_kernel(const float* __restrict__ x,      // [256][14]
                     const float* __restrict__ w,      // [6][14][3]
                     float* __restrict__ out)          // [256][14]
{
    extern __shared__ unsigned char smem_raw[];
    float2* psi = reinterpret_cast<float2*>(smem_raw);  // 16384 x float2 = 128 KB

    const unsigned tid = threadIdx.x;
    const unsigned b   = blockIdx.x;

    // |0...0>
    for (unsigned s = tid; s < NSTATES; s += NTHREADS)
        psi[s] = make_float2((s == 0u) ? 1.f : 0.f, 0.f);
    __syncthreads();

    // ---- Angle embedding: RX(x_i * pi) on wire i ----------------------------
    for (int i = 0; i < NQ; ++i) {
        const float half = x[b * NQ + i] * 3.14159265358979323846f * 0.5f;
        float sh, ch;
        sincosf(half, &sh, &ch);
        // U = [[c, -i s], [-i s, c]]
        apply_1q_wmma(psi, i,
                      ch, 0.f,   0.f, -sh,
                      0.f, -sh,  ch, 0.f);
        __syncthreads();
    }

    // ---- Variational layers -------------------------------------------------
    for (int l = 0; l < NL; ++l) {
        // Rot(phi, theta, omega) = RZ(omega) RY(theta) RZ(phi) on every wire
        for (int i = 0; i < NQ; ++i) {
            const float* wl   = &w[(l * NQ + i) * 3];
            const float phi   = wl[0];
            const float theta = wl[1];
            const float omg   = wl[2];
            float st, ct, sa, ca, sb, cb;
            sincosf(0.5f * theta,        &st, &ct);
            sincosf(0.5f * (phi + omg),  &sa, &ca);   // ep = exp(-i a)
            sincosf(0.5f * (phi - omg),  &sb, &cb);   // em = exp(-i b)
            // u00 = ep*ct, u01 = -conj(em)*st, u10 = em*st, u11 = conj(ep)*ct
            apply_1q_wmma(psi, i,
                          ct * ca, -ct * sa,
                          -st * cb, -st * sb,
                          st * cb, -st * sb,
                          ct * ca,  ct * sa);
            __syncthreads();
        }
        // CNOT ring: control i, target (i+1)%NQ
        for (int i = 0; i < NQ; ++i) {
            const int c   = i;
            const int t   = (i + 1) % NQ;
            const int sc  = 13 - c;
            const int stt = 13 - t;
            const int pHi = (sc > stt) ? sc : stt;
            const int pLo = (sc > stt) ? stt : sc;
            // 4096 swap pairs: states with bit_c = 1; swap target bit 0 <-> 1
#pragma unroll 2
            for (unsigned q = tid; q < 4096u; q += NTHREADS) {
                unsigned v = q;
                v = ((v >> pLo) << (pLo + 1)) | (v & ((1u << pLo) - 1u));
                v = ((v >> pHi) << (pHi + 1)) | (v & ((1u << pHi) - 1u));
                const unsigned i0 = v | (1u << sc);   // control=1, target=0
                const unsigned i1 = i0 | (1u << stt); // control=1, target=1
                const float2 a0 = psi[i0];
                const float2 a1 = psi[i1];
                psi[i0] = a1;
                psi[i1] = a0;
            }
            __syncthreads();
        }
    }

    // ---- <Z_i> = sum_s |psi_s|^2 * (1 - 2*bit_i(s)) -------------------------
    float acc[NQ];
#pragma unroll
    for (int i = 0; i < NQ; ++i) acc[i] = 0.f;
    for (unsigned s = tid; s < NSTATES; s += NTHREADS) {
        const float2 v  = psi[s];
        const float  pr = v.x * v.x + v.y * v.y;
#pragma unroll
        for (int i = 0; i < NQ; ++i)
            acc[i] += ((s >> (13 - i)) & 1u) ? -pr : pr;
    }
    __syncthreads();   // psi dead from here; reuse its LDS as reduction scratch

    float* scratch = reinterpret_cast<float*>(smem_raw);  // [NQ][NTHREADS]
    for (int i = 0; i < NQ; ++i)
        scratch[i * NTHREADS + tid] = acc[i];
    __syncthreads();

    // Deterministic fixed-order reduction (bit-reproducible across replays)
    if (tid < NQ) {
        float sum = 0.f;
        for (int t2 = 0; t2 < NTHREADS; ++t2)
            sum += scratch[tid * NTHREADS + t2];
        out[b * NQ + tid] = sum;
    }
}

extern "C" void kernel_launch(void* const* d_in, const int* in_sizes, int n_in,
                              void* d_out, int out_size, void* d_ws, size_t ws_size,
                              hipStream_t stream) {
    (void)in_sizes; (void)n_in; (void)out_size; (void)d_ws; (void)ws_size;
    const float* x = (const float*)d_in[0];   // [256][14] float32
    const float* w = (const float*)d_in[1];   // [6][14][3] float32
    float* out     = (float*)d_out;           // [256][14] float32

    const size_t shmem = (size_t)NSTATES * sizeof(float2);   // 131072 B dynamic LDS
    // Opt in to >64KB dynamic LDS where the runtime requires it (ignore rc).
    (void)hipFuncSetAttribute((const void*)qvl_qsim_kernel,
                              hipFuncAttributeMaxDynamicSharedMemorySize,
                              (int)shmem);

    qvl_qsim_kernel<<<256, NTHREADS, shmem, stream>>>(x, w, out);
}